// RegressionLoss_59622736003394
// MI455X (gfx1250) — compile-verified
//
#include <hip/hip_runtime.h>
#include <hip/hip_bf16.h>
#include <math.h>

// Problem constants (match reference)
#define NB 16
#define NA 65536
#define NM 64
#define TPB 256
#define CHUNKS (NA / TPB)      // 256 blocks per image
#define NWAVES (TPB / 32)      // 8 waves per block (wave32)

typedef __attribute__((ext_vector_type(2))) float v2f;
typedef __attribute__((ext_vector_type(8))) float v8f;

// Reduce (lossVal, nposVal) across the 32 lanes of a wave using the fp32
// tensor path: D = A(16x4) x B(4x16), wave32.
// A layout (32-bit 16x4): lanes 0-15 -> M=lane, VGPR0=K0, VGPR1=K1;
//                         lanes 16-31 -> M=lane-16, VGPR0=K2, VGPR1=K3.
// We put loss in VGPR0 (columns K0/K2) and npos in VGPR1 (columns K1/K3).
// B selector: output cols 0-7 sum K0+K2 (loss), cols 8-15 sum K1+K3 (npos).
// After summing the 8 D VGPRs per lane and folding the M-halves with a
// shfl_xor(16), lane 0 holds total loss and lane 8 holds total npos.
__device__ __forceinline__ void wave_reduce2_wmma(float lossVal, float nposVal,
                                                  float& lossTot, float& nposTot) {
  const unsigned lane = threadIdx.x & 31u;
  const unsigned n = lane & 15u;
  v2f a; a.x = lossVal; a.y = nposVal;
  v2f b; b.x = (n < 8u) ? 1.0f : 0.0f;
         b.y = (n < 8u) ? 0.0f : 1.0f;
  v8f c = {};
  c = __builtin_amdgcn_wmma_f32_16x16x4_f32(false, a, false, b,
                                            (short)0, c, false, false);
  float s = c[0] + c[1] + c[2] + c[3] + c[4] + c[5] + c[6] + c[7];
  s += __shfl_xor(s, 16, 32);          // fold M=0..7 with M=8..15
  lossTot = __shfl(s, 0, 32);
  nposTot = __shfl(s, 8, 32);
}

// Kernel 1: one thread per (image, anchor). Boxes for the image live in LDS
// as float4 {b0, b1, area, 0}; invalid boxes encoded as [+1e30,-1e30] so
// their IoU clamps to exactly 0 (never >= 0.5, never beats a positive IoU).
// IoU max/argmax tracked divide-free via cross-multiplied fractions.
__global__ void __launch_bounds__(TPB)
regloss_partials(const float* __restrict__ reg,      // (B, A, 2)
                 const float* __restrict__ anchors,  // (1, A, 2)
                 const float* __restrict__ ann,      // (B, M, 3)
                 float* __restrict__ pLoss,          // (B*CHUNKS)
                 float* __restrict__ pN) {           // (B*CHUNKS)
  __shared__ float4 sbox[NM];
  __shared__ float wL[NWAVES], wN[NWAVES];

  const int bb    = blockIdx.x;
  const int img   = bb >> 8;          // / CHUNKS
  const int chunk = bb & (CHUNKS - 1);
  const int tid   = threadIdx.x;

  if (tid < NM) {
    const float* p = ann + ((size_t)img * NM + tid) * 3;
    float b0 = p[0], b1 = p[1];
    bool valid = (p[2] != -1.0f);
    if (!valid) { b0 = 1e30f; b1 = -1e30f; }
    sbox[tid] = make_float4(b0, b1, b1 - b0, 0.0f);
  }
  __syncthreads();

  const int aidx = chunk * TPB + tid;
  const float2 anc = ((const float2*)anchors)[aidx];
  const float aw   = anc.y - anc.x;
  const float actr = anc.x + 0.5f * aw;

  // best IoU as fraction bi/bu ; init to -1 so index 0 always wins first.
  float bi = -1.0f, bu = 1.0f;
  int barg = 0;
  #pragma unroll 4
  for (int m = 0; m < NM; ++m) {
    const float4 bx = sbox[m];
    float iw = fminf(anc.y, bx.y) - fmaxf(anc.x, bx.x);
    iw = fmaxf(iw, 0.0f);
    const float ua = fmaxf(aw + bx.z - iw, 1e-8f);
    // iw/ua > bi/bu  <=>  iw*bu > bi*ua   (ua, bu > 0); strict > keeps first
    if (iw * bu > bi * ua) { bi = iw; bu = ua; barg = m; }
  }
  const bool pos = (2.0f * bi >= bu);  // iou_max >= 0.5

  const float2 r = ((const float2*)reg)[(size_t)img * NA + aidx];
  const float4 g = sbox[barg];
  float gw   = g.y - g.x;
  float gctr = g.x + 0.5f * gw;
  gw = fmaxf(gw, 1.0f);
  const float tdx = (gctr - actr) / aw / 0.1f;
  const float tdw = logf(gw / aw) / 0.2f;

  const float third = 1.0f / 9.0f;
  const float d0 = fabsf(tdx - r.x);
  const float d1 = fabsf(tdw - r.y);
  const float l0 = (d0 <= third) ? 4.5f * d0 * d0 : d0 - (0.5f / 9.0f);
  const float l1 = (d1 <= third) ? 4.5f * d1 * d1 : d1 - (0.5f / 9.0f);

  const float lossPart = pos ? (l0 + l1) : 0.0f;
  const float nposPart = pos ? 1.0f : 0.0f;

  float lt, nt;
  wave_reduce2_wmma(lossPart, nposPart, lt, nt);
  const int wid = tid >> 5;
  if ((tid & 31) == 0) { wL[wid] = lt; wN[wid] = nt; }
  __syncthreads();
  if (tid == 0) {
    float L = 0.0f, N = 0.0f;
    #pragma unroll
    for (int w = 0; w < NWAVES; ++w) { L += wL[w]; N += wN[w]; }
    pLoss[bb] = L;
    pN[bb]    = N;
  }
}

// Kernel 2: single block reduces the 256 partials of each image (fixed
// order -> deterministic), applies per-image normalization + validity
// masks, and writes the mean over images.
__global__ void __launch_bounds__(TPB)
regloss_finalize(const float* __restrict__ ann,
                 const float* __restrict__ pLoss,
                 const float* __restrict__ pN,
                 float* __restrict__ out) {
  __shared__ float wL[NWAVES], wN[NWAVES];
  __shared__ int sValid[NB];
  const int tid = threadIdx.x;

  if (tid < NB) sValid[tid] = 0;
  __syncthreads();
  for (int i = tid; i < NB * NM; i += TPB) {
    if (ann[(size_t)i * 3 + 2] != -1.0f) atomicOr(&sValid[i / NM], 1);
  }
  __syncthreads();

  float acc = 0.0f;
  for (int img = 0; img < NB; ++img) {
    float lt, nt;
    wave_reduce2_wmma(pLoss[img * CHUNKS + tid], pN[img * CHUNKS + tid], lt, nt);
    const int wid = tid >> 5;
    if ((tid & 31) == 0) { wL[wid] = lt; wN[wid] = nt; }
    __syncthreads();
    if (tid == 0) {
      float L = 0.0f, N = 0.0f;
      #pragma unroll
      for (int w = 0; w < NWAVES; ++w) { L += wL[w]; N += wN[w]; }
      float loss = L / (2.0f * fmaxf(N, 1.0f));
      if (!(N > 0.0f) || sValid[img] == 0) loss = 0.0f;
      acc += loss;
    }
    __syncthreads();
  }
  if (tid == 0) out[0] = acc / (float)NB;
}

extern "C" void kernel_launch(void* const* d_in, const int* in_sizes, int n_in,
                              void* d_out, int out_size, void* d_ws, size_t ws_size,
                              hipStream_t stream) {
  const float* reg     = (const float*)d_in[0];  // (16, 65536, 2) f32
  const float* anchors = (const float*)d_in[1];  // (1, 65536, 2) f32
  const float* ann     = (const float*)d_in[2];  // (16, 64, 3) f32
  float* out = (float*)d_out;

  float* pLoss = (float*)d_ws;                   // NB*CHUNKS floats
  float* pN    = pLoss + (size_t)NB * CHUNKS;    // NB*CHUNKS floats (32 KiB total)

  regloss_partials<<<NB * CHUNKS, TPB, 0, stream>>>(reg, anchors, ann, pLoss, pN);
  regloss_finalize<<<1, TPB, 0, stream>>>(ann, pLoss, pN, out);
}